// M3_56727928045684
// MI455X (gfx1250) — compile-verified
//
#include <hip/hip_runtime.h>
#include <math.h>

// ---------------------------------------------------------------------------
// Problem constants (from reference): B=4, H=W=64, C=128 -> L=4096, DI=256,
// NS=16, R=8, K=4 (causal conv), S=4 streams.
// ---------------------------------------------------------------------------
namespace {
constexpr int B_  = 4;
constexpr int H_  = 64;
constexpr int W_  = 64;
constexpr int C_  = 128;
constexpr int L_  = H_ * W_;          // 4096
constexpr int DI_ = 2 * C_;           // 256
constexpr int NS_ = 16;
constexpr int R_  = 8;
constexpr int K_  = 4;
constexpr int S_  = 4;
constexpr long BL_ = (long)B_ * L_;   // 16384
constexpr int P_  = R_ + 2 * NS_;     // 40 (xproj output width)
}

typedef float v2f __attribute__((ext_vector_type(2)));
typedef float v8f __attribute__((ext_vector_type(8)));

__device__ __forceinline__ float wred_sum(float v) {
#pragma unroll
  for (int o = 16; o > 0; o >>= 1) v += __shfl_xor(v, o, 32);
  return v;
}
__device__ __forceinline__ float wred_max(float v) {
#pragma unroll
  for (int o = 16; o > 0; o >>= 1) v = fmaxf(v, __shfl_xor(v, o, 32));
  return v;
}
__device__ __forceinline__ float sigmoidf_(float x) { return 1.f / (1.f + __expf(-x)); }
__device__ __forceinline__ float siluf_(float x)    { return x * sigmoidf_(x); }

// ---------------------------------------------------------------------------
// Generic batched GEMM on the CDNA5 matrix pipe:
//   Cmat[M x N] = A[M x K] @ Wt[N x K]^T      (computes X @ W^T)
// One wave32 owns a 16 x 64 strip (4 N-tiles): a single A fragment feeds 4
// V_WMMA_F32_16X16X4_F32 issues into 4 independent accumulators (covers the
// WMMA->WMMA RAW hazard and quarters A traffic). Ragged N handled with a
// clamped row address + mask multiply -- no EXEC manipulation in the loop.
// A fragment (16x4): lanes 0-15 hold K=k0,k0+1 ; lanes 16-31 hold K=k0+2,k0+3
// B fragment (4x16): mirrored (N on lanes).  C/D: 8 VGPRs, M=r(+8), N=lane%16.
// ---------------------------------------------------------------------------
__global__ __launch_bounds__(256) void k_gemm_wmma_f32(
    const float* __restrict__ A, const float* __restrict__ Wt,
    float* __restrict__ Cmat, int M, int N, int K,
    long aBatch, long wBatch, long cBatch) {
  const int lane = threadIdx.x & 31;
  const int wave = threadIdx.x >> 5;
  const int mtiles  = M >> 4;
  const int ntiles  = (N + 15) >> 4;
  const int ngroups = (ntiles + 3) >> 2;
  const int tile = blockIdx.x * 8 + wave;
  if (tile >= mtiles * ngroups) return;
  const int mt = tile / ngroups;
  const int ng = tile % ngroups;

  const float* Ab = A  + (long)blockIdx.z * aBatch;
  const float* Wb = Wt + (long)blockIdx.z * wBatch;
  float*       Cb = Cmat + (long)blockIdx.z * cBatch;

  const int ridx  = lane & 15;          // M-row (A) / N-col (B) within tile
  const int khalf = (lane >> 4) << 1;   // 0 or 2
  const long m = (long)mt * 16 + ridx;

  // Per-strip column bookkeeping: clamp to a valid row, zero via mask.
  const float* wrow[4];
  float nmask[4];
#pragma unroll
  for (int j = 0; j < 4; ++j) {
    const int n  = (ng * 4 + j) * 16 + ridx;
    const int nc = (n < N) ? n : (N - 1);
    nmask[j] = (n < N) ? 1.f : 0.f;
    wrow[j]  = Wb + (long)nc * K + khalf;
  }

  v8f acc[4] = {};
  const float* ap = Ab + m * K + khalf;
  for (int k0 = 0; k0 < K; k0 += 4) {
    v2f a;
    a.x = ap[k0];
    a.y = ap[k0 + 1];
    __builtin_prefetch(ap + k0 + 4, 0, 3);  // gfx1250 global_prefetch, next K-block
#pragma unroll
    for (int j = 0; j < 4; ++j) {
      v2f b;
      b.x = wrow[j][k0] * nmask[j];
      b.y = wrow[j][k0 + 1] * nmask[j];
      acc[j] = __builtin_amdgcn_wmma_f32_16x16x4_f32(
          /*neg_a=*/false, a, /*neg_b=*/false, b,
          /*c_mod=*/(short)0, acc[j], /*reuse_a=*/false, /*reuse_b=*/false);
    }
  }

  const long cmr = (long)mt * 16 + ((lane >> 4) << 3); // M = r + (lane>=16 ? 8 : 0)
#pragma unroll
  for (int j = 0; j < 4; ++j) {
    const int cn = (ng * 4 + j) * 16 + (lane & 15);
    if (cn < N) {
#pragma unroll
      for (int r = 0; r < 8; ++r) Cb[(cmr + r) * N + cn] = acc[j][r];
    }
  }
}

// ---------------------------------------------------------------------------
// Double LayerNorm over C=128 per (b,l) row. One wave32 per row, 4 channels
// per lane, reductions via __shfl_xor (wave32-native, no LDS).
// Optional: Xadd (residual add, raw sum written to outRaw) and per-row scale.
// ---------------------------------------------------------------------------
__global__ __launch_bounds__(256) void k_ln2(
    const float* __restrict__ X, const float* __restrict__ Xadd,
    const float* __restrict__ scale,
    const float* __restrict__ w, const float* __restrict__ bv,
    float* __restrict__ outLn2, float* __restrict__ outRaw) {
  const int wave = threadIdx.x >> 5, lane = threadIdx.x & 31;
  const long row  = (long)blockIdx.x * 8 + wave;
  const long base = row * C_ + lane * 4;
  float v[4];
#pragma unroll
  for (int i = 0; i < 4; ++i) v[i] = X[base + i];
  if (Xadd) {
#pragma unroll
    for (int i = 0; i < 4; ++i) v[i] += Xadd[base + i];
  }
  if (outRaw) {
#pragma unroll
    for (int i = 0; i < 4; ++i) outRaw[base + i] = v[i];
  }
  if (scale) {
    const float s = scale[row];
#pragma unroll
    for (int i = 0; i < 4; ++i) v[i] *= s;
  }
  float wv[4], bb[4];
#pragma unroll
  for (int i = 0; i < 4; ++i) { wv[i] = w[lane * 4 + i]; bb[i] = bv[lane * 4 + i]; }
#pragma unroll
  for (int pass = 0; pass < 2; ++pass) {
    float s = v[0] + v[1] + v[2] + v[3];
    s = wred_sum(s);
    const float mu = s * (1.f / C_);
    float q = 0.f;
#pragma unroll
    for (int i = 0; i < 4; ++i) { const float d = v[i] - mu; q += d * d; }
    q = wred_sum(q);
    const float inv = rsqrtf(q * (1.f / C_) + 1e-5f);
#pragma unroll
    for (int i = 0; i < 4; ++i) v[i] = (v[i] - mu) * inv * wv[i] + bb[i];
  }
#pragma unroll
  for (int i = 0; i < 4; ++i) outLn2[base + i] = v[i];
}

// mean/max over C per (b,l) row -> mm[b, 0..1, L]
__global__ __launch_bounds__(256) void k_spmap(const float* __restrict__ X,
                                               float* __restrict__ mm) {
  const int wave = threadIdx.x >> 5, lane = threadIdx.x & 31;
  const long row  = (long)blockIdx.x * 8 + wave;
  const long base = row * C_ + lane * 4;
  float s = 0.f, m = -3.402823466e38f;
#pragma unroll
  for (int i = 0; i < 4; ++i) { const float x = X[base + i]; s += x; m = fmaxf(m, x); }
  s = wred_sum(s);
  m = wred_max(m);
  if (lane == 0) {
    const int b = (int)(row >> 12), l = (int)(row & (L_ - 1));
    mm[(long)b * 2 * L_ + l]      = s * (1.f / C_);
    mm[(long)b * 2 * L_ + L_ + l] = m;
  }
}

// 5x5 conv (pad 2) over the 2-channel mean/max map -> sigmoid gate sc[b*L+l]
__global__ __launch_bounds__(256) void k_spconv(const float* __restrict__ mm,
                                                const float* __restrict__ spw,
                                                float* __restrict__ sc) {
  const long idx = (long)blockIdx.x * 256 + threadIdx.x; // b*L + l
  const int b = (int)(idx >> 12), l = (int)(idx & (L_ - 1));
  const int y = l >> 6, x = l & 63;
  float o = 0.f;
  for (int ic = 0; ic < 2; ++ic) {
    const float* mp = mm + ((long)b * 2 + ic) * L_;
    for (int ky = 0; ky < 5; ++ky) {
      const int yy = y + ky - 2;
      if (yy < 0 || yy >= H_) continue;
      for (int kx = 0; kx < 5; ++kx) {
        const int xx = x + kx - 2;
        if (xx < 0 || xx >= W_) continue;
        o += mp[yy * W_ + xx] * spw[(ic * 5 + ky) * 5 + kx];
      }
    }
  }
  sc[idx] = sigmoidf_(o);
}

// Causal depthwise conv (K=4) + SiLU over the 4 streams. Stream 1 reads the
// L-reversed x-slice of xz directly (no materialized reversed copy).
__global__ __launch_bounds__(256) void k_conv_silu(
    const float* __restrict__ xz, const float* __restrict__ u2,
    const float* __restrict__ u3, const float* __restrict__ cw,
    const float* __restrict__ cbb, float* __restrict__ uc) {
  const long idx = (long)blockIdx.x * 256 + threadIdx.x; // ((s*B+b)*L+l)*DI + d
  const int d = (int)(idx & (DI_ - 1));
  const int l = (int)((idx >> 8) & (L_ - 1));
  const int b = (int)((idx >> 20) & 3);
  const int s = (int)(idx >> 22);
  float acc = cbb[s * DI_ + d];
#pragma unroll
  for (int k = 0; k < K_; ++k) {
    const int ls = l + k - (K_ - 1);
    if (ls < 0) continue;
    float uval;
    if (s == 0)      uval = xz[((long)b * L_ + ls) * (2 * DI_) + d];
    else if (s == 1) uval = xz[((long)b * L_ + (L_ - 1 - ls)) * (2 * DI_) + d];
    else if (s == 2) uval = u2[((long)b * L_ + ls) * DI_ + d];
    else             uval = u3[((long)b * L_ + ls) * DI_ + d];
    acc += cw[(s * DI_ + d) * K_ + k] * uval;
  }
  uc[idx] = siluf_(acc);
}

// Selective scan: one block per (s,b), one thread per channel d. NS=16 state
// registers per thread; dt-proj + softplus fused. The 40-float proj row for
// step l is staged straight into LDS with the gfx1250 async data mover path
// (global_load_async_to_lds_b32, ASYNCcnt), then published with
// s_wait_asynccnt 0 + barrier. Stream 1 stores its output pre-reversed.
__global__ __launch_bounds__(256) void k_scan(
    const float* __restrict__ proj, const float* __restrict__ uc,
    const float* __restrict__ dtw, const float* __restrict__ dtb,
    const float* __restrict__ alog, const float* __restrict__ dsk,
    float* __restrict__ y) {
  __shared__ float pbuf[P_];
  const int s = blockIdx.x >> 2, b = blockIdx.x & 3;
  const int d = threadIdx.x;
  const int sd = s * DI_ + d;
  float wrow[R_];
#pragma unroll
  for (int r = 0; r < R_; ++r) wrow[r] = dtw[sd * R_ + r];
  const float bias = dtb[sd];
  float Am[NS_];
#pragma unroll
  for (int n = 0; n < NS_; ++n) Am[n] = -__expf(alog[sd * NS_ + n]);
  const float Dk = dsk[sd];
  float h[NS_];
#pragma unroll
  for (int n = 0; n < NS_; ++n) h[n] = 0.f;
  const long pbase = ((long)s * B_ + b) * L_ * P_;
  const long ubase = (((long)s * B_ + b) * L_) * DI_;
#if defined(__gfx1250__)
  // Generic LDS pointers carry the raw LDS byte offset in their low 32 bits.
  const unsigned lds_off =
      (threadIdx.x < P_) ? (unsigned)(size_t)(&pbuf[threadIdx.x]) : 0u;
#endif
  for (int l = 0; l < L_; ++l) {
    if (threadIdx.x < P_) {
#if defined(__gfx1250__)
      const unsigned long long ga =
          (unsigned long long)(proj + pbase + (long)l * P_ + threadIdx.x);
      asm volatile("global_load_async_to_lds_b32 %0, %1, off"
                   :: "v"(lds_off), "v"(ga) : "memory");
#else
      pbuf[threadIdx.x] = proj[pbase + (long)l * P_ + threadIdx.x];
#endif
    }
#if defined(__gfx1250__)
    asm volatile("s_wait_asynccnt 0" ::: "memory");
#endif
    __syncthreads();
    float dtp = bias;
#pragma unroll
    for (int r = 0; r < R_; ++r) dtp += pbuf[r] * wrow[r];
    const float dt = (dtp > 20.f) ? dtp : log1pf(__expf(dtp));
    const float u  = uc[ubase + (long)l * DI_ + d];
    const float du = dt * u;
    float ys = 0.f;
#pragma unroll
    for (int n = 0; n < NS_; ++n) {
      h[n] = h[n] * __expf(dt * Am[n]) + du * pbuf[R_ + n];
      ys += h[n] * pbuf[R_ + NS_ + n];
    }
    const int lst = (s == 1) ? (L_ - 1 - l) : l;
    y[ubase + (long)lst * DI_ + d] = ys + u * Dk;
    __syncthreads();
  }
}

// g = (y0+y1+y2+y3) * silu(z)
__global__ __launch_bounds__(256) void k_combine(const float* __restrict__ y,
                                                 const float* __restrict__ xz,
                                                 float* __restrict__ g) {
  const long idx = (long)blockIdx.x * 256 + threadIdx.x; // (b*L+l)*DI + d
  const long stride = BL_ * DI_;
  const float s = y[idx] + y[idx + stride] + y[idx + 2 * stride] + y[idx + 3 * stride];
  const long row = idx >> 8;
  const int  d   = (int)(idx & (DI_ - 1));
  const float z  = xz[row * (2 * DI_) + DI_ + d];
  g[idx] = s * siluf_(z);
}

// Channel-attention reductions: one wave per (b,c): mean & max over L
__global__ __launch_bounds__(256) void k_ca_reduce(const float* __restrict__ gf,
                                                   float* __restrict__ cm,
                                                   float* __restrict__ cx) {
  const int wid  = blockIdx.x * 8 + (threadIdx.x >> 5);
  const int lane = threadIdx.x & 31;
  const int b = wid >> 7, c = wid & (C_ - 1);
  float s = 0.f, m = -3.402823466e38f;
  for (int l = lane; l < L_; l += 32) {
    const float v = gf[((long)b * L_ + l) * C_ + c];
    s += v;
    m = fmaxf(m, v);
  }
  s = wred_sum(s);
  m = wred_max(m);
  if (lane == 0) { cm[b * C_ + c] = s * (1.f / L_); cx[b * C_ + c] = m; }
}

// sig = sigmoid(fc(mean) + fc(max)); fc: C->32 relu ->C (tiny, one block per b)
__global__ __launch_bounds__(128) void k_ca_fc(const float* __restrict__ cm,
                                               const float* __restrict__ cx,
                                               const float* __restrict__ w1,
                                               const float* __restrict__ w2,
                                               float* __restrict__ sig) {
  __shared__ float sm[C_], sx[C_], hm[32], hx[32];
  const int b = blockIdx.x, t = threadIdx.x;
  sm[t] = cm[b * C_ + t];
  sx[t] = cx[b * C_ + t];
  __syncthreads();
  if (t < 32) {
    float am = 0.f, ax = 0.f;
    for (int c = 0; c < C_; ++c) {
      const float wv = w1[t * C_ + c];
      am += wv * sm[c];
      ax += wv * sx[c];
    }
    hm[t] = fmaxf(am, 0.f);
    hx[t] = fmaxf(ax, 0.f);
  }
  __syncthreads();
  float o = 0.f;
  for (int j = 0; j < 32; ++j) o += w2[t * 32 + j] * (hm[j] + hx[j]);
  sig[b * C_ + t] = sigmoidf_(o);
}

// t[b,c,l] = sig[b,c] * gf[b,l,c]^2   (transposes to NCHW for the dw conv)
__global__ __launch_bounds__(256) void k_sqsig(const float* __restrict__ gf,
                                               const float* __restrict__ sig,
                                               float* __restrict__ t) {
  const long idx = (long)blockIdx.x * 256 + threadIdx.x; // (b*L+l)*C + c
  const int  c  = (int)(idx & (C_ - 1));
  const long bl = idx >> 7;
  const int  b  = (int)(bl >> 12);
  const int  l  = (int)(bl & (L_ - 1));
  const float v = gf[idx];
  t[((long)b * C_ + c) * L_ + l] = v * v * sig[b * C_ + c];
}

// 3x3 depthwise conv (pad 1) + bias + residual, write final out[b,l,c]
__global__ __launch_bounds__(256) void k_dwconv(const float* __restrict__ t,
                                                const float* __restrict__ dww,
                                                const float* __restrict__ dwb,
                                                float* __restrict__ out) {
  const long idx = (long)blockIdx.x * 256 + threadIdx.x; // (b*C+c)*L + l
  const int  l  = (int)(idx & (L_ - 1));
  const long bc = idx >> 12;
  const int  c  = (int)(bc & (C_ - 1));
  const int  b  = (int)(bc >> 7);
  const int  y  = l >> 6, x = l & 63;
  const float* tp = t + bc * L_;
  float o = 0.f;
#pragma unroll
  for (int ky = 0; ky < 3; ++ky) {
    const int yy = y + ky - 1;
    if (yy < 0 || yy >= H_) continue;
#pragma unroll
    for (int kx = 0; kx < 3; ++kx) {
      const int xx = x + kx - 1;
      if (xx < 0 || xx >= W_) continue;
      o += tp[yy * W_ + xx] * dww[c * 9 + ky * 3 + kx];
    }
  }
  o += dwb[c] + tp[l];
  out[((long)b * L_ + l) * C_ + c] = o;
}

// ---------------------------------------------------------------------------
extern "C" void kernel_launch(void* const* d_in, const int* in_sizes, int n_in,
                              void* d_out, int out_size, void* d_ws, size_t ws_size,
                              hipStream_t stream) {
  (void)in_sizes; (void)n_in; (void)out_size; (void)ws_size;
  const float* I1   = (const float*)d_in[0];
  const float* fres = (const float*)d_in[1];
  const float* I2   = (const float*)d_in[2];
  const float* fus  = (const float*)d_in[3];
  const float* ln1w = (const float*)d_in[4];
  const float* ln1b = (const float*)d_in[5];
  const float* ln2w = (const float*)d_in[6];
  const float* ln2b = (const float*)d_in[7];
  const float* ln3w = (const float*)d_in[8];
  const float* ln3b = (const float*)d_in[9];
  const float* spw  = (const float*)d_in[10];
  const float* caw1 = (const float*)d_in[11];
  const float* caw2 = (const float*)d_in[12];
  const float* dww  = (const float*)d_in[13];
  const float* dwb  = (const float*)d_in[14];
  const float* ipw  = (const float*)d_in[15];
  const float* ip1w = (const float*)d_in[16];
  const float* ip2w = (const float*)d_in[17];
  const float* cw   = (const float*)d_in[18];
  const float* cbb  = (const float*)d_in[19];
  const float* xpw  = (const float*)d_in[20];
  const float* dtw  = (const float*)d_in[21];
  const float* dtb  = (const float*)d_in[22];
  const float* alog = (const float*)d_in[23];
  const float* dsk  = (const float*)d_in[24];
  const float* outw = (const float*)d_in[25];

  const long BLC = BL_ * C_; // 2,097,152
  float* ws = (float*)d_ws;
  float* xf = ws;
  float* e1 = xf + BLC;
  float* e2 = e1 + BLC;
  float* mm = e2 + BLC;                    // B*2*L
  float* sc = mm + (long)B_ * 2 * L_;      // B*L
  float* xz = sc + BL_;                    // B*L*512
  float* u2 = xz + BL_ * 2 * DI_;          // B*L*256
  float* u3 = u2 + BL_ * DI_;
  float* uc = u3 + BL_ * DI_;              // S*B*L*256
  float* pr = uc + (long)S_ * BL_ * DI_;   // S*B*L*40
  float* yb = pr + (long)S_ * BL_ * P_;    // S*B*L*256
  float* g  = yb + (long)S_ * BL_ * DI_;   // B*L*256
  float* gf = g + BL_ * DI_;               // B*L*128
  float* tt = gf + BLC;                    // B*C*L
  float* cm = tt + BLC;
  float* cx = cm + B_ * C_;
  float* sg = cx + B_ * C_;

  float* out0  = (float*)d_out;            // out (B,L,C)
  float* frout = out0 + BLC;               // fusion_resi (B,L,C)

  auto launch_gemm = [&](const float* A, const float* Wt, float* Cm,
                         int M, int N, int Kd, int batch,
                         long ab, long wb, long cb) {
    const int mt = M / 16;
    const int nt = (N + 15) / 16;
    const int ng = (nt + 3) / 4;             // 4 N-tiles per wave
    const int tiles = mt * ng;
    dim3 grid((tiles + 7) / 8, 1, batch);
    k_gemm_wmma_f32<<<grid, 256, 0, stream>>>(A, Wt, Cm, M, N, Kd, ab, wb, cb);
  };

  // fusion_resi = fusion + fusion_resi (written to output); xf = ln(ln(fr))
  k_ln2<<<BL_ / 8, 256, 0, stream>>>(fus, fres, nullptr, ln1w, ln1b, xf, frout);
  // e1 = ln(ln(I2 * sp_att(I2)))  (ln2 weights)
  k_spmap<<<BL_ / 8, 256, 0, stream>>>(I2, mm);
  k_spconv<<<(int)(BL_ / 256), 256, 0, stream>>>(mm, spw, sc);
  k_ln2<<<BL_ / 8, 256, 0, stream>>>(I2, nullptr, sc, ln2w, ln2b, e1, nullptr);
  // e2 = ln(ln(I1 * sp_att(I1)))  (ln3 weights)
  k_spmap<<<BL_ / 8, 256, 0, stream>>>(I1, mm);
  k_spconv<<<(int)(BL_ / 256), 256, 0, stream>>>(mm, spw, sc);
  k_ln2<<<BL_ / 8, 256, 0, stream>>>(I1, nullptr, sc, ln3w, ln3b, e2, nullptr);

  // Projections on the WMMA pipe
  launch_gemm(xf, ipw, xz, (int)BL_, 2 * DI_, C_, 1, 0, 0, 0); // xz = xf @ in_proj^T
  launch_gemm(e1, ip1w, u2, (int)BL_, DI_, C_, 1, 0, 0, 0);
  launch_gemm(e2, ip2w, u3, (int)BL_, DI_, C_, 1, 0, 0, 0);

  // Causal dwconv + SiLU -> uc (stream 1 reads x reversed)
  k_conv_silu<<<(int)((long)S_ * BL_ * DI_ / 256), 256, 0, stream>>>(xz, u2, u3, cw, cbb, uc);

  // proj[s] = uc[s] @ xproj_w[s]^T  (batched over S)
  launch_gemm(uc, xpw, pr, (int)BL_, P_, DI_, S_,
              BL_ * DI_, (long)P_ * DI_, BL_ * P_);

  // Selective scan (dt + softplus fused); stream 1 output stored reversed
  k_scan<<<S_ * B_, 256, 0, stream>>>(pr, uc, dtw, dtb, alog, dsk, yb);

  // g = sum_s y * silu(z); gf = g @ out_w^T
  k_combine<<<(int)(BL_ * DI_ / 256), 256, 0, stream>>>(yb, xz, g);
  launch_gemm(g, outw, gf, (int)BL_, C_, DI_, 1, 0, 0, 0);

  // Channel attention + square gate + depthwise 3x3 + residual -> out
  k_ca_reduce<<<(B_ * C_) / 8, 256, 0, stream>>>(gf, cm, cx);
  k_ca_fc<<<B_, 128, 0, stream>>>(cm, cx, caw1, caw2, sg);
  k_sqsig<<<(int)(BLC / 256), 256, 0, stream>>>(gf, sg, tt);
  k_dwconv<<<(int)(BLC / 256), 256, 0, stream>>>(tt, dww, dwb, out0);
}